// Memory_15436112462037
// MI455X (gfx1250) — compile-verified
//
#include <hip/hip_runtime.h>

typedef float v2f __attribute__((ext_vector_type(2)));
typedef float v8f __attribute__((ext_vector_type(8)));

namespace {
constexpr int B_SZ = 64, U_SZ = 16, D_SZ = 1024, N_SZ = 2048, O_SZ = 1024;
constexpr int KC   = 32;   // K-slab staged in LDS per iteration
constexpr int ASTR = 36;   // padded LDS row stride for the 64xKC A tile
// B tile kept k-pair interleaved in LDS: lB[(k>>1)*256 + n*2 + (k&1)]
// -> each WMMA B fragment {B[k][n], B[k+1][n]} is one aligned ds_load_b64.
}

// ---------------------------------------------------------------------------
// Kernel 1: logits[b,u] = dot(X[b,u,:], adaptive_lr[u,:]) / temperature
// ---------------------------------------------------------------------------
__global__ __launch_bounds__(256) void logits_kernel(
    const float* __restrict__ X, const float* __restrict__ alr,
    const float* __restrict__ temperature, float* __restrict__ logits) {
  const int gw   = (blockIdx.x * 256 + threadIdx.x) >> 5;
  const int lane = threadIdx.x & 31;
  const int b = gw / U_SZ, u = gw % U_SZ;
  const float* xp = X + (size_t)(b * U_SZ + u) * D_SZ;
  const float* ap = alr + (size_t)u * D_SZ;
  float s = 0.f;
  for (int i = lane; i < D_SZ; i += 32) s += xp[i] * ap[i];
#pragma unroll
  for (int off = 16; off; off >>= 1) s += __shfl_down(s, off, 32);
  if (lane == 0) logits[b * U_SZ + u] = s / temperature[0];
}

// ---------------------------------------------------------------------------
// Kernel 2: lr[b,:] = softmax over units of logits[b,:]
// ---------------------------------------------------------------------------
__global__ void softmax_units_kernel(const float* __restrict__ logits,
                                     float* __restrict__ lr) {
  const int b = threadIdx.x;
  if (b >= B_SZ) return;
  float v[U_SZ];
  float m = -3.4e38f;
#pragma unroll
  for (int u = 0; u < U_SZ; ++u) { v[u] = logits[b * U_SZ + u]; m = fmaxf(m, v[u]); }
  float s = 0.f;
#pragma unroll
  for (int u = 0; u < U_SZ; ++u) { v[u] = __expf(v[u] - m); s += v[u]; }
  const float inv = 1.f / s;
#pragma unroll
  for (int u = 0; u < U_SZ; ++u) lr[b * U_SZ + u] = v[u] * inv;
}

// ---------------------------------------------------------------------------
// shuffle two global row-pair float4s into the interleaved LDS format
// ---------------------------------------------------------------------------
__device__ __forceinline__ void store_bpair(float* __restrict__ lB, int rp,
                                            int c4, float4 g0, float4 g1) {
  const int base = rp * 256 + c4 * 2;   // 16-byte aligned (c4 % 4 == 0)
  float4 s0, s1;
  s0.x = g0.x; s0.y = g1.x; s0.z = g0.y; s0.w = g1.y;
  s1.x = g0.z; s1.y = g1.z; s1.z = g0.w; s1.w = g1.w;
  *(float4*)&lB[base]     = s0;
  *(float4*)&lB[base + 4] = s1;
}

// ---------------------------------------------------------------------------
// Kernel 3: fused  acc = X@Win + state@(W*sr)
//           new_state = (1-lr)*state + lr*tanh(acc + bias)
// block = 8 waves, tile = 64(B) x 128(N); wave = 16(B) x 64(N) = 4 WMMA accums
// ---------------------------------------------------------------------------
__global__ __launch_bounds__(256) void state_update_kernel(
    const float* __restrict__ X, const float* __restrict__ state,
    const float* __restrict__ W, const float* __restrict__ Win,
    const float* __restrict__ bias, const float* __restrict__ sr,
    const float* __restrict__ lr, float* __restrict__ new_state) {
  __shared__ float lA[64 * ASTR];
  __shared__ float lB[(KC / 2) * 256];

  const int tid  = threadIdx.x;
  const int u    = blockIdx.x >> 4;           // 16 n-tiles per unit
  const int n0   = (blockIdx.x & 15) * 128;
  const int wave = tid >> 5, lane = tid & 31;
  const int wm = wave & 3, wn = wave >> 2;    // 4 M-waves x 2 N-waves
  const int hi = lane >> 4, l15 = lane & 15;
  const float srv = sr[u];

  const int aRow = tid >> 3, aCol = (tid & 7) * 4;  // A: 64x32 tile, 2 passes
  const int rp0  = tid >> 5;                        // B row-pair 0..7 (+8 for pass 2)
  const int c4   = (tid & 31) * 4;                  // B col quad

  v8f acc[4] = {};
  float4 ra[2], rb[4];

  const int FEED_IT = D_SZ / KC;           // 32 iterations over X/Win
  const int TOT     = FEED_IT + N_SZ / KC; // + 64 iterations over state/W

  // stage tile 0 (feed phase, k0 = 0) into registers
#pragma unroll
  for (int j = 0; j < 2; ++j)
    ra[j] = *(const float4*)(X + (size_t)((aRow + 32 * j) * U_SZ + u) * D_SZ + aCol);
#pragma unroll
  for (int j = 0; j < 2; ++j) {   // work item j: row-pair rp0 + 8*j
    const int k = 2 * (rp0 + 8 * j);
    rb[2 * j]     = *(const float4*)(Win + (size_t)(u * D_SZ + k) * N_SZ + n0 + c4);
    rb[2 * j + 1] = *(const float4*)(Win + (size_t)(u * D_SZ + k + 1) * N_SZ + n0 + c4);
  }

  for (int it = 0; it < TOT; ++it) {
    __syncthreads();
#pragma unroll
    for (int j = 0; j < 2; ++j)
      *(float4*)&lA[(aRow + 32 * j) * ASTR + aCol] = ra[j];
#pragma unroll
    for (int j = 0; j < 2; ++j)
      store_bpair(lB, rp0 + 8 * j, c4, rb[2 * j], rb[2 * j + 1]);
    __syncthreads();

    const int nx = it + 1;
    if (nx < TOT) {                     // prefetch next K-slab (overlaps WMMAs)
      if (nx < FEED_IT) {
        const int k0 = nx * KC;
#pragma unroll
        for (int j = 0; j < 2; ++j)
          ra[j] = *(const float4*)(X + (size_t)((aRow + 32 * j) * U_SZ + u) * D_SZ + k0 + aCol);
#pragma unroll
        for (int j = 0; j < 2; ++j) {
          const int k = k0 + 2 * (rp0 + 8 * j);
          rb[2 * j]     = *(const float4*)(Win + (size_t)(u * D_SZ + k) * N_SZ + n0 + c4);
          rb[2 * j + 1] = *(const float4*)(Win + (size_t)(u * D_SZ + k + 1) * N_SZ + n0 + c4);
        }
      } else {
        const int k0 = (nx - FEED_IT) * KC;
#pragma unroll
        for (int j = 0; j < 2; ++j)
          ra[j] = *(const float4*)(state + (size_t)((aRow + 32 * j) * U_SZ + u) * N_SZ + k0 + aCol);
#pragma unroll
        for (int j = 0; j < 2; ++j) {
          const int k = k0 + 2 * (rp0 + 8 * j);
          float4 t0 = *(const float4*)(W + (size_t)(u * N_SZ + k) * N_SZ + n0 + c4);
          float4 t1 = *(const float4*)(W + (size_t)(u * N_SZ + k + 1) * N_SZ + n0 + c4);
          t0.x *= srv; t0.y *= srv; t0.z *= srv; t0.w *= srv;  // fold sr into W
          t1.x *= srv; t1.y *= srv; t1.z *= srv; t1.w *= srv;
          rb[2 * j] = t0; rb[2 * j + 1] = t1;
        }
      }
      // L2 prefetch of the dominant (weight) stream a few slabs ahead
      const int pf = it + 3;
      if (pf < TOT) {
        const float* p = (pf < FEED_IT)
            ? Win + (size_t)(u * D_SZ + pf * KC + 2 * rp0) * N_SZ + n0 + c4
            : W   + (size_t)(u * N_SZ + (pf - FEED_IT) * KC + 2 * rp0) * N_SZ + n0 + c4;
        __builtin_prefetch(p, 0, 1);
      }
    }

#pragma unroll
    for (int kc = 0; kc < KC; kc += 4) {
      // A fragment 16x4: lane m = l15, K = vgpr + 2*hi  -> ds_load_b64
      const float* ap = &lA[(wm * 16 + l15) * ASTR + kc + hi * 2];
      v2f a; a.x = ap[0]; a.y = ap[1];
      // B fragment 4x16: rows (kc+2*hi, +1), col = wn*64 + t*16 + l15
      const int brow = kc + hi * 2;  // even
      const float* bp = &lB[(brow >> 1) * 256 + (wn * 64 + l15) * 2];
#pragma unroll
      for (int t = 0; t < 4; ++t) {
        v2f bb; bb.x = bp[t * 32]; bb.y = bp[t * 32 + 1];   // one ds_load_b64
        acc[t] = __builtin_amdgcn_wmma_f32_16x16x4_f32(
            false, a, false, bb, (short)0, acc[t], false, false);
      }
    }
  }

  // epilogue: C/D layout -> b = wm*16 + v + 8*hi, n = n0 + wn*64 + t*16 + l15
  float lrv[8];
#pragma unroll
  for (int v = 0; v < 8; ++v)
    lrv[v] = lr[(wm * 16 + v + hi * 8) * U_SZ + u];

#pragma unroll
  for (int t = 0; t < 4; ++t) {
    const int n = n0 + wn * 64 + t * 16 + l15;
    const float bv = bias[u * N_SZ + n];
#pragma unroll
    for (int v = 0; v < 8; ++v) {
      const int b = wm * 16 + v + hi * 8;
      const size_t idx = (size_t)(b * U_SZ + u) * N_SZ + n;
      const float th = tanhf(acc[t][v] + bv);
      const float sv = state[idx];
      new_state[idx] = (1.f - lrv[v]) * sv + lrv[v] * th;
    }
  }
}

// ---------------------------------------------------------------------------
// Kernel 4: output = new_state @ Wout   (per unit: 64x2048 @ 2048x1024)
// ---------------------------------------------------------------------------
__global__ __launch_bounds__(256) void output_kernel(
    const float* __restrict__ NS, const float* __restrict__ Wout,
    float* __restrict__ out) {
  __shared__ float lA[64 * ASTR];
  __shared__ float lB[(KC / 2) * 256];

  const int tid  = threadIdx.x;
  const int u    = blockIdx.x >> 3;           // 8 o-tiles per unit
  const int o0   = (blockIdx.x & 7) * 128;
  const int wave = tid >> 5, lane = tid & 31;
  const int wm = wave & 3, wn = wave >> 2;
  const int hi = lane >> 4, l15 = lane & 15;

  const int aRow = tid >> 3, aCol = (tid & 7) * 4;
  const int rp0  = tid >> 5;
  const int c4   = (tid & 31) * 4;

  v8f acc[4] = {};
  float4 ra[2], rb[4];
  const int TOT = N_SZ / KC;  // 64

#pragma unroll
  for (int j = 0; j < 2; ++j)
    ra[j] = *(const float4*)(NS + (size_t)((aRow + 32 * j) * U_SZ + u) * N_SZ + aCol);
#pragma unroll
  for (int j = 0; j < 2; ++j) {
    const int k = 2 * (rp0 + 8 * j);
    rb[2 * j]     = *(const float4*)(Wout + (size_t)(u * N_SZ + k) * O_SZ + o0 + c4);
    rb[2 * j + 1] = *(const float4*)(Wout + (size_t)(u * N_SZ + k + 1) * O_SZ + o0 + c4);
  }

  for (int it = 0; it < TOT; ++it) {
    __syncthreads();
#pragma unroll
    for (int j = 0; j < 2; ++j)
      *(float4*)&lA[(aRow + 32 * j) * ASTR + aCol] = ra[j];
#pragma unroll
    for (int j = 0; j < 2; ++j)
      store_bpair(lB, rp0 + 8 * j, c4, rb[2 * j], rb[2 * j + 1]);
    __syncthreads();

    const int nx = it + 1;
    if (nx < TOT) {
      const int k0 = nx * KC;
#pragma unroll
      for (int j = 0; j < 2; ++j)
        ra[j] = *(const float4*)(NS + (size_t)((aRow + 32 * j) * U_SZ + u) * N_SZ + k0 + aCol);
#pragma unroll
      for (int j = 0; j < 2; ++j) {
        const int k = k0 + 2 * (rp0 + 8 * j);
        rb[2 * j]     = *(const float4*)(Wout + (size_t)(u * N_SZ + k) * O_SZ + o0 + c4);
        rb[2 * j + 1] = *(const float4*)(Wout + (size_t)(u * N_SZ + k + 1) * O_SZ + o0 + c4);
      }
      const int pf = it + 3;
      if (pf < TOT)
        __builtin_prefetch(
            Wout + (size_t)(u * N_SZ + pf * KC + 2 * rp0) * O_SZ + o0 + c4, 0, 1);
    }

#pragma unroll
    for (int kc = 0; kc < KC; kc += 4) {
      const float* ap = &lA[(wm * 16 + l15) * ASTR + kc + hi * 2];
      v2f a; a.x = ap[0]; a.y = ap[1];
      const int brow = kc + hi * 2;
      const float* bp = &lB[(brow >> 1) * 256 + (wn * 64 + l15) * 2];
#pragma unroll
      for (int t = 0; t < 4; ++t) {
        v2f bb; bb.x = bp[t * 32]; bb.y = bp[t * 32 + 1];
        acc[t] = __builtin_amdgcn_wmma_f32_16x16x4_f32(
            false, a, false, bb, (short)0, acc[t], false, false);
      }
    }
  }

#pragma unroll
  for (int t = 0; t < 4; ++t) {
    const int o = o0 + wn * 64 + t * 16 + l15;
#pragma unroll
    for (int v = 0; v < 8; ++v) {
      const int b = wm * 16 + v + hi * 8;
      out[(size_t)(b * U_SZ + u) * O_SZ + o] = acc[t][v];
    }
  }
}

// ---------------------------------------------------------------------------
extern "C" void kernel_launch(void* const* d_in, const int* in_sizes, int n_in,
                              void* d_out, int out_size, void* d_ws, size_t ws_size,
                              hipStream_t stream) {
  const float* X     = (const float*)d_in[0];
  const float* state = (const float*)d_in[1];
  const float* W     = (const float*)d_in[2];
  const float* Win   = (const float*)d_in[3];
  const float* bias  = (const float*)d_in[4];
  const float* Wout  = (const float*)d_in[5];
  const float* alr   = (const float*)d_in[6];
  const float* sr    = (const float*)d_in[7];
  const float* temp  = (const float*)d_in[8];

  float* new_state = (float*)d_out;                               // [B,U,N]
  float* output    = (float*)d_out + (size_t)B_SZ * U_SZ * N_SZ;  // [B,U,O]

  float* logits = (float*)d_ws;              // B*U floats
  float* lr     = logits + B_SZ * U_SZ;      // B*U floats

  logits_kernel<<<(B_SZ * U_SZ) / 8, 256, 0, stream>>>(X, alr, temp, logits);
  softmax_units_kernel<<<1, 64, 0, stream>>>(logits, lr);
  state_update_kernel<<<U_SZ * (N_SZ / 128), 256, 0, stream>>>(
      X, state, W, Win, bias, sr, lr, new_state);
  output_kernel<<<U_SZ * (O_SZ / 128), 256, 0, stream>>>(new_state, Wout, output);
}